// Mamba_2757369004784
// MI455X (gfx1250) — compile-verified
//
#include <hip/hip_runtime.h>
#include <math.h>

typedef __attribute__((ext_vector_type(2))) float v2f;
typedef __attribute__((ext_vector_type(8))) float v8f;

#define GEMM_WAVES 8   // waves per block
#define GEMM_NT    4   // 16-wide N tiles per wave
#define GEMM_MT    2   // 16-tall M tiles per wave -> wave tile = 32(M) x 64(N)

// ---------------------------------------------------------------------------
// Generic fp32 WMMA GEMM:  C[m, n] = sum_k A[m, k] * W[n, k]   (+ epilogue)
//   m = b*Lseq + l  (M = B*Lseq rows, grid.y = M/32)
//   A element address  = A + b*a_bs + l*a_ls + k        (k contiguous)
//   W is row-major (N, K)
//   C element address  = C + b*c_bs + l*c_ls + n*c_ns
//   mode 0: plain store; mode 1: softplus(acc + bias[n])  (dt_proj)
// ---------------------------------------------------------------------------
__global__ __launch_bounds__(256) void wmma_gemm_f32(
    const float* __restrict__ A, const float* __restrict__ W,
    float* __restrict__ C, const float* __restrict__ bias,
    int Lseq, int N, int K,
    long a_bs, long a_ls,
    long c_bs, long c_ls, long c_ns,
    int mode)
{
    const int lane = threadIdx.x & 31;
    const int wave = threadIdx.x >> 5;
    const int hi   = lane >> 4;    // half-wave: 0 or 1
    const int lo   = lane & 15;
    const int kk   = 2 * hi;       // this half-wave's K-pair offset within a chunk

    const int m_tile = blockIdx.y * (GEMM_MT * 16);
    const int nb     = (blockIdx.x * GEMM_WAVES + wave) * (GEMM_NT * 16);

    // A-fragment rows (16x4 layout: lanes 0-15 = M rows, halves split K pairs)
    const int  m0 = m_tile + lo;
    const int  m1 = m0 + 16;
    const long a_off0 = (long)(m0 / Lseq) * a_bs + (long)(m0 % Lseq) * a_ls;
    const long a_off1 = (long)(m1 / Lseq) * a_bs + (long)(m1 % Lseq) * a_ls;

    // B-fragment rows (weights, row-major NxK); clamp OOB columns, stores guarded
    long w_off[GEMM_NT];
    int  ncol[GEMM_NT];
    #pragma unroll
    for (int t = 0; t < GEMM_NT; ++t) {
        int n   = nb + t * 16 + lo;
        ncol[t] = n;
        w_off[t] = (long)((n < N) ? n : (N - 1)) * (long)K;
    }

    v8f acc0[GEMM_NT] = {};
    v8f acc1[GEMM_NT] = {};

    // ---- main loop: no bounds checks, 4 K-chunks (K=16) per iteration ----
    const int Kmain = K & ~15;
    for (int kb = 0; kb < Kmain; kb += 16) {
        // speculative prefetch ~4 iterations ahead (safe unguarded)
        __builtin_prefetch(&A[a_off0 + kb + 64], 0, 1);
        __builtin_prefetch(&A[a_off1 + kb + 64], 0, 1);
        __builtin_prefetch(&W[w_off[0] + kb + 64], 0, 1);
        __builtin_prefetch(&W[w_off[3] + kb + 64], 0, 1);
        #pragma unroll
        for (int c = 0; c < 4; ++c) {
            const int k0 = kb + c * 4 + kk;
            v2f a0, a1;
            a0.x = A[a_off0 + k0];
            a0.y = A[a_off0 + k0 + 1];
            a1.x = A[a_off1 + k0];
            a1.y = A[a_off1 + k0 + 1];
            #pragma unroll
            for (int t = 0; t < GEMM_NT; ++t) {
                v2f bf;
                bf.x = W[w_off[t] + k0];
                bf.y = W[w_off[t] + k0 + 1];
                acc0[t] = __builtin_amdgcn_wmma_f32_16x16x4_f32(
                    false, a0, false, bf, (short)0, acc0[t], false, false);
                acc1[t] = __builtin_amdgcn_wmma_f32_16x16x4_f32(
                    false, a1, false, bf, (short)0, acc1[t], false, false);
            }
        }
    }

    // ---- tail: only runs when K % 16 != 0 (in_proj K=1025) ----
    for (int kb = Kmain; kb < K; kb += 4) {
        const int k0 = kb + kk;
        v2f a0, a1;
        a0.x = (k0     < K) ? A[a_off0 + k0]     : 0.0f;
        a0.y = (k0 + 1 < K) ? A[a_off0 + k0 + 1] : 0.0f;
        a1.x = (k0     < K) ? A[a_off1 + k0]     : 0.0f;
        a1.y = (k0 + 1 < K) ? A[a_off1 + k0 + 1] : 0.0f;
        #pragma unroll
        for (int t = 0; t < GEMM_NT; ++t) {
            v2f bf;
            bf.x = (k0     < K) ? W[w_off[t] + k0]     : 0.0f;
            bf.y = (k0 + 1 < K) ? W[w_off[t] + k0 + 1] : 0.0f;
            acc0[t] = __builtin_amdgcn_wmma_f32_16x16x4_f32(
                false, a0, false, bf, (short)0, acc0[t], false, false);
            acc1[t] = __builtin_amdgcn_wmma_f32_16x16x4_f32(
                false, a1, false, bf, (short)0, acc1[t], false, false);
        }
    }

    // ---- store; C/D layout: VGPR r = row (tile + r + 8*hi), col = lane&15 ----
    #pragma unroll
    for (int mt = 0; mt < GEMM_MT; ++mt) {
        const int  m_base = m_tile + mt * 16 + 8 * hi;
        const long b2     = m_base / Lseq;
        const int  l_base = m_base % Lseq;
        #pragma unroll
        for (int t = 0; t < GEMM_NT; ++t) {
            const int n = ncol[t];
            if (n >= N) continue;
            const float bv = (mode == 1) ? bias[n] : 0.0f;
            #pragma unroll
            for (int r = 0; r < 8; ++r) {
                float v = (mt == 0) ? acc0[t][r] : acc1[t][r];
                if (mode == 1) {                         // softplus(v + b_dt)
                    v += bv;
                    v = (v > 20.0f) ? v : log1pf(__expf(v));
                }
                C[b2 * c_bs + (long)(l_base + r) * c_ls + (long)n * c_ns] = v;
            }
        }
    }
}

// ---------------------------------------------------------------------------
// Depthwise causal conv1d (width 4, left pad 3) + SiLU.
// xz laid out (b, l, 4096); channels [0,2048) are x. Output xc (b, l, 2048).
// ---------------------------------------------------------------------------
__global__ __launch_bounds__(256) void conv_silu_kernel(
    const float* __restrict__ xz, const float* __restrict__ cw,
    const float* __restrict__ cb, float* __restrict__ xc,
    int Lseq, int Din, int E, long total)
{
    long idx = (long)blockIdx.x * blockDim.x + threadIdx.x;
    if (idx >= total) return;
    const int  d   = (int)(idx % Din);
    const long bl  = idx / Din;          // b*L + l
    const int  l   = (int)(bl % Lseq);
    const long base = bl * (long)E + d;  // (b, l, d) inside xz
    float acc = cb[d];
    #pragma unroll
    for (int j = 0; j < 4; ++j) {
        const int ls = l - 3 + j;
        if (ls >= 0) acc += xz[base + (long)(j - 3) * E] * cw[d * 4 + j];
    }
    xc[idx] = acc / (1.0f + __expf(-acc));   // SiLU
}

// ---------------------------------------------------------------------------
// Selective scan. One thread per (b, d); 16 states in registers.
// B_t / C_t (32 floats, x_dbl channels 64..95) double-buffered in LDS via
// CDNA5 async global->LDS copies: wave 0 issues global_load_async_to_lds_b32
// for step l+1 at the top of step l, then s_wait_asynccnt 0 + barrier at the
// bottom, hiding the global latency behind the 16-wide exp/fma chain.
// Fuses D*u skip and SiLU(z) gate: y = (scan + D*u) * silu(z).
// ---------------------------------------------------------------------------
__global__ __launch_bounds__(256) void scan_kernel(
    const float* __restrict__ delta, const float* __restrict__ xc,
    const float* __restrict__ xdbl,  const float* __restrict__ xz,
    const float* __restrict__ A_log, const float* __restrict__ Dp,
    float* __restrict__ y, int Lseq, int Din)
{
    __shared__ float sBC[2][32];
    const int tid  = threadIdx.x;
    const int blkD = Din / 256;                   // blocks per batch
    const int b    = blockIdx.x / blkD;
    const int d    = (blockIdx.x % blkD) * 256 + tid;

    float Arow[16], h[16];
    #pragma unroll
    for (int n = 0; n < 16; ++n) {
        Arow[n] = -__expf(A_log[d * 16 + n]);
        h[n]    = 0.0f;
    }
    const float Dv   = Dp[d];
    const long  row0 = (long)b * Lseq;
    const long  E    = 2L * Din;

    // prologue: async-stage B/C for l = 0 into buffer 0 (wave 0 only)
    if (tid < 32) {
        unsigned           dst = (unsigned)(unsigned long long)&sBC[0][tid];
        unsigned long long src = (unsigned long long)(xdbl + row0 * 96 + 64 + tid);
        asm volatile("global_load_async_to_lds_b32 %0, %1, off"
                     :: "v"(dst), "v"(src) : "memory");
        asm volatile("s_wait_asynccnt 0x0" ::: "memory");
    }
    __syncthreads();

    for (int l = 0; l < Lseq; ++l) {
        const int buf = l & 1;
        const long bl = row0 + l;

        // kick off async copy of next step's B/C into the other buffer
        if (tid < 32 && (l + 1) < Lseq) {
            unsigned           dst = (unsigned)(unsigned long long)&sBC[buf ^ 1][tid];
            unsigned long long src = (unsigned long long)(xdbl + (bl + 1) * 96 + 64 + tid);
            asm volatile("global_load_async_to_lds_b32 %0, %1, off"
                         :: "v"(dst), "v"(src) : "memory");
        }

        const float dt = delta[bl * Din + d];
        const float u  = xc[bl * Din + d];
        const float du = dt * u;
        float yv = 0.0f;
        #pragma unroll
        for (int n = 0; n < 16; ++n) {
            h[n] = h[n] * __expf(dt * Arow[n]) + du * sBC[buf][n];
            yv  += h[n] * sBC[buf][16 + n];
        }
        const float z = xz[bl * E + Din + d];
        const float g = z / (1.0f + __expf(-z));            // silu(z)
        y[bl * Din + d] = (yv + Dv * u) * g;

        // drain this wave's async copy, then publish LDS to all waves
        if (tid < 32) {
            asm volatile("s_wait_asynccnt 0x0" ::: "memory");
        }
        __syncthreads();
    }
}

// ---------------------------------------------------------------------------
extern "C" void kernel_launch(void* const* d_in, const int* in_sizes, int n_in,
                              void* d_out, int out_size, void* d_ws, size_t ws_size,
                              hipStream_t stream) {
    (void)in_sizes; (void)n_in; (void)out_size; (void)ws_size;
    const float* hs     = (const float*)d_in[0];  // (2, 2048, 1025)
    const float* W_in   = (const float*)d_in[1];  // (4096, 1025)
    const float* conv_w = (const float*)d_in[2];  // (2048, 4)
    const float* conv_b = (const float*)d_in[3];  // (2048)
    const float* W_x    = (const float*)d_in[4];  // (96, 2048)
    const float* W_dt   = (const float*)d_in[5];  // (2048, 64)
    const float* b_dt   = (const float*)d_in[6];  // (2048)
    const float* A_log  = (const float*)d_in[7];  // (2048, 16)
    const float* Dp     = (const float*)d_in[8];  // (2048)
    const float* W_out  = (const float*)d_in[9];  // (1024, 2048)
    float* out = (float*)d_out;                   // (2, 2048, 1024)

    const int Bt = 2, L = 2048, Dm = 1024, Din = 2048, E = 4096, Dh = 1025, R = 96;

    // Workspace layout (all fp32, (b, l, channel)); ~162 MB — L2 resident.
    float* xz   = (float*)d_ws;                    // (B, L, 4096)  x | z
    float* xc   = xz   + (size_t)Bt * L * E;       // (B, L, 2048)  conv+silu(x)
    float* xdbl = xc   + (size_t)Bt * L * Din;     // (B, L, 96)    dt_lo | B | C
    float* delt = xdbl + (size_t)Bt * L * R;       // (B, L, 2048)  softplus dt
    float* yb   = delt + (size_t)Bt * L * Din;     // (B, L, 2048)  gated scan out

    const dim3 blk(256, 1, 1);
    const int  Mt = Bt * L / (GEMM_MT * 16);       // 128 M-tiles (32 rows each)
    const int  NW = GEMM_WAVES * GEMM_NT * 16;     // 512 N-cols per block

    // 1) in_proj: xz[b,l,e] = hs[b,l,:] . W_in[e,:]
    wmma_gemm_f32<<<dim3((E + NW - 1) / NW, Mt), blk, 0, stream>>>(
        hs, W_in, xz, nullptr, L, E, Dh,
        (long)L * Dh, (long)Dh,
        (long)L * E, (long)E, 1L, 0);

    // 2) depthwise causal conv + SiLU on x half
    const long totc = (long)Bt * L * Din;
    conv_silu_kernel<<<(unsigned)((totc + 255) / 256), blk, 0, stream>>>(
        xz, conv_w, conv_b, xc, L, Din, E, totc);

    // 3) x_proj: xdbl[b,l,e] = xc[b,l,:] . W_x[e,:]
    wmma_gemm_f32<<<dim3((R + NW - 1) / NW, Mt), blk, 0, stream>>>(
        xc, W_x, xdbl, nullptr, L, R, Din,
        (long)L * Din, (long)Din,
        (long)L * R, (long)R, 1L, 0);

    // 4) dt_proj + softplus: delt[b,l,d] = softplus(xdbl[b,l,0:64] . W_dt[d,:] + b_dt[d])
    wmma_gemm_f32<<<dim3((Din + NW - 1) / NW, Mt), blk, 0, stream>>>(
        xdbl, W_dt, delt, b_dt, L, Din, 64,
        (long)L * R, (long)R,
        (long)L * Din, (long)Din, 1L, 1);

    // 5) selective scan + D-skip + SiLU(z) gate (async LDS double-buffered B/C)
    scan_kernel<<<Bt * (Din / 256), blk, 0, stream>>>(
        delt, xc, xdbl, xz, A_log, Dp, yb, L, Din);

    // 6) out_proj: out[b,l,o] = yb[b,l,:] . W_out[o,:]
    wmma_gemm_f32<<<dim3((Dm + NW - 1) / NW, Mt), blk, 0, stream>>>(
        yb, W_out, out, nullptr, L, Dm, Din,
        (long)L * Din, (long)Din,
        (long)L * Dm, (long)Dm, 1L, 0);
}